// LengthPredictor_28217935135403
// MI455X (gfx1250) — compile-verified
//
#include <hip/hip_runtime.h>

// Problem constants (from reference)
#define B_        32
#define L_        2048
#define E_        1024
#define OUT_SIZE_ 41
#define MAX_TGT_  (L_ + 20)   // 2068
#define NSPLIT    8           // L-split for maxpool parallelism

typedef __attribute__((ext_vector_type(2))) float v2f;
typedef __attribute__((ext_vector_type(4))) float v4f;
typedef __attribute__((ext_vector_type(8))) float v8f;

__device__ __forceinline__ void nt_store_v4(v4f* p, v4f v) {
    // Non-temporal 128b store: don't pollute L2 (we need L2 capacity for x).
    __builtin_nontemporal_store(v, p);
}

__device__ __forceinline__ v4f vmax4(v4f a, v4f b) {
    v4f r;
    r.x = fmaxf(a.x, b.x);
    r.y = fmaxf(a.y, b.y);
    r.z = fmaxf(a.z, b.z);
    r.w = fmaxf(a.w, b.w);
    return r;
}

// ---------------------------------------------------------------------------
// Kernel 1: partial max over a slice of L.
// grid = B_*NSPLIT blocks, 256 threads; each thread owns one v4f column
// (256 threads * 4 floats == E_). Fully coalesced b128 loads.
// Loads of x use default (RT) caching on purpose: x is re-read by the gather
// pass and 192MB L2 can hold most of the 256MB tensor.
// ---------------------------------------------------------------------------
__global__ void pool_partial_kernel(const float* __restrict__ x,
                                    float* __restrict__ partial) {
    const int blk   = blockIdx.x;
    const int b     = blk / NSPLIT;
    const int split = blk % NSPLIT;
    const int tid   = threadIdx.x;           // v4f column index, 0..255

    const int rows = L_ / NSPLIT;            // 256
    const int l0   = split * rows;

    const v4f* xp = (const v4f*)(x + (size_t)b * L_ * E_)
                    + (size_t)l0 * (E_ / 4) + tid;

    v4f m = xp[0];
    for (int l = 1; l < rows; ++l) {
        m = vmax4(m, xp[(size_t)l * (E_ / 4)]);
    }

    v4f* pp = (v4f*)(partial + ((size_t)b * NSPLIT + split) * E_);
    nt_store_v4(&pp[tid], m);
}

// ---------------------------------------------------------------------------
// Kernel 2: reduce the NSPLIT partials -> pooled (B_ x E_).
// grid = B_ blocks, 256 threads (v4f per thread).
// ---------------------------------------------------------------------------
__global__ void pool_reduce_kernel(const float* __restrict__ partial,
                                   float* __restrict__ pooled) {
    const int b   = blockIdx.x;
    const int tid = threadIdx.x;

    const v4f* pp = (const v4f*)(partial + (size_t)b * NSPLIT * E_);
    v4f m = pp[tid];
    for (int sp = 1; sp < NSPLIT; ++sp) {
        m = vmax4(m, pp[(size_t)sp * (E_ / 4) + tid]);
    }
    ((v4f*)(pooled + (size_t)b * E_))[tid] = m;
}

// ---------------------------------------------------------------------------
// Kernel 3: len_out_prob = pooled(32x1024) @ W(1024x41) + bias, via
// V_WMMA_F32_16X16X4_F32 (full fp32 matrix pipe).
// One block of 6 waves; wave w handles tile (mt = w/3, nt = w%3), K-loop of
// 256 WMMAs.
//
// VGPR layouts (ISA 7.12.2):
//   A (16x4):  lane m = lane&15; VGPR v holds K = 2*(lane>=16) + v
//   B (4x16):  lane n = lane&15; VGPR v holds K = 2*(lane>=16) + v
//   C/D (16x16): VGPR v, lane half h: M = v + 8*h, N = lane&15
// ---------------------------------------------------------------------------
__global__ void len_gemm_wmma_kernel(const float* __restrict__ pooled,
                                     const float* __restrict__ W,
                                     const float* __restrict__ bias,
                                     float* __restrict__ len_out) {
    const int tid  = threadIdx.x;
    const int wave = tid >> 5;
    const int lane = tid & 31;
    const int half = lane >> 4;        // 0 or 1
    const int lm   = lane & 15;

    const int mt = wave / 3;           // 0..1  (M tile)
    const int nt = wave % 3;           // 0..2  (N tile)

    const int m = mt * 16 + lm;        // < 32 always
    const int n = nt * 16 + lm;        // may exceed 40 on last tile
    const int nvalid = (n < OUT_SIZE_);
    const int nc = nvalid ? n : (OUT_SIZE_ - 1);   // clamped address (no branch)

    v8f acc = {};

    const float* arow = pooled + (size_t)m * E_;

    for (int k = 0; k < E_; k += 4) {
        const int kb = k + 2 * half;

        v2f a;
        a.x = arow[kb + 0];
        a.y = arow[kb + 1];

        v2f bm;
        {
            float b0 = W[(size_t)(kb + 0) * OUT_SIZE_ + nc];
            float b1 = W[(size_t)(kb + 1) * OUT_SIZE_ + nc];
            bm.x = nvalid ? b0 : 0.0f;
            bm.y = nvalid ? b1 : 0.0f;
        }

        // (neg_a, A, neg_b, B, c_mod, C, reuse_a, reuse_b)
        acc = __builtin_amdgcn_wmma_f32_16x16x4_f32(
            false, a, false, bm, (short)0, acc, false, false);
    }

    if (nvalid) {
        const float bv = bias[n];
#pragma unroll
        for (int v = 0; v < 8; ++v) {
            const int mo = mt * 16 + v + 8 * half;   // < 32
            len_out[(size_t)mo * OUT_SIZE_ + n] = acc[v] + bv;
        }
    }
}

// ---------------------------------------------------------------------------
// Kernel 4: gather rows: out[b, j, :] = x[b, idx(b,j), :]
// grid = B_*MAX_TGT_ blocks, 256 threads: one 4 KiB row per block (v4f).
// Output is written once and never re-read -> non-temporal stores so the
// 260MB stream doesn't evict x from the 192MB L2 (x rows are re-read many
// times when tgt_length > src_length).
// ---------------------------------------------------------------------------
__global__ void gather_rows_kernel(const float* __restrict__ x,
                                   const int* __restrict__ src_len,
                                   const int* __restrict__ tgt_len,
                                   float* __restrict__ out) {
    const int blk = blockIdx.x;
    const int b   = blk / MAX_TGT_;
    const int j   = blk % MAX_TGT_;

    const int s = src_len[b];
    const int t = tgt_len[b];
    // j*s <= 2067*2048 ~ 4.2M : fits int32. Floor-div == trunc-div (non-neg).
    const int idx = (j < t) ? ((j * s) / t)
                            : ((s < (L_ - 1)) ? s : (L_ - 1));

    const v4f* src = (const v4f*)(x + ((size_t)b * L_ + idx) * E_);
    v4f*       dst = (v4f*)(out + ((size_t)b * MAX_TGT_ + j) * E_);
    v4f v = src[threadIdx.x];
    nt_store_v4(&dst[threadIdx.x], v);
}

// ---------------------------------------------------------------------------
// Kernel 5: tgt_length passthrough (int -> float value).
// ---------------------------------------------------------------------------
__global__ void write_tgt_kernel(const int* __restrict__ tgt,
                                 float* __restrict__ out) {
    const int i = threadIdx.x;
    if (i < B_) out[i] = (float)tgt[i];
}

// ---------------------------------------------------------------------------
extern "C" void kernel_launch(void* const* d_in, const int* in_sizes, int n_in,
                              void* d_out, int out_size, void* d_ws, size_t ws_size,
                              hipStream_t stream) {
    const float* x       = (const float*)d_in[0];  // (B, L, E) f32
    const float* W       = (const float*)d_in[1];  // (E, OUT_SIZE) f32
    const float* bias    = (const float*)d_in[2];  // (OUT_SIZE,) f32
    const int*   src_len = (const int*)d_in[3];    // (B,) i32
    const int*   tgt_len = (const int*)d_in[4];    // (B,) i32

    float* out = (float*)d_out;
    // Output layout: [output (B*MAX_TGT*E)] [len_out_prob (B*OUT_SIZE)] [tgt (B)]
    float* out_gather = out;
    float* out_prob   = out + (size_t)B_ * MAX_TGT_ * E_;
    float* out_tgt    = out_prob + (size_t)B_ * OUT_SIZE_;

    // Workspace: partial maxes (B*NSPLIT*E f32 = 1 MiB) + pooled (B*E = 128 KiB)
    float* partial = (float*)d_ws;
    float* pooled  = partial + (size_t)B_ * NSPLIT * E_;

    pool_partial_kernel<<<B_ * NSPLIT, 256, 0, stream>>>(x, partial);
    pool_reduce_kernel<<<B_, 256, 0, stream>>>(partial, pooled);
    len_gemm_wmma_kernel<<<1, 192, 0, stream>>>(pooled, W, bias, out_prob);
    gather_rows_kernel<<<B_ * MAX_TGT_, 256, 0, stream>>>(x, src_len, tgt_len, out_gather);
    write_tgt_kernel<<<1, 32, 0, stream>>>(tgt_len, out_tgt);
}